// MultiboxLoss_57578331571031
// MI455X (gfx1250) — compile-verified
//
#include <hip/hip_runtime.h>
#include <hip/hip_bf16.h>

#define B_ 64
#define P_ 24564
#define C_ 81
#define NOBJ_ 32
#define TILES_ 1536            // ceil(P_/16)
#define LN81_ 4.3944491546724391f
#define NEGINF_ -3.402823466e+38f

typedef float v8f __attribute__((ext_vector_type(8)));
typedef float v2f __attribute__((ext_vector_type(2)));

__device__ __forceinline__ unsigned long long packKey(float o, int p) {
    return (((unsigned long long)__float_as_uint(o)) << 32) |
           (unsigned long long)(0xFFFFFFFFu - (unsigned)p);
}

// ---------------- init: zero accumulators / keys ----------------
__global__ void k_init(unsigned long long* gkey, int* numPos, float* acc) {
    int i = blockIdx.x * blockDim.x + threadIdx.x;
    if (i < B_ * NOBJ_) gkey[i] = 0ull;
    if (i < B_) numPos[i] = 0;
    if (i < 2) acc[i] = 0.f;
}

// ---------------- matching: per-prior best truth + per-truth best prior ----------------
__global__ void __launch_bounds__(256) k_match(const float* __restrict__ tgt,
                                               const float* __restrict__ anc,
                                               signed char* __restrict__ mt,
                                               unsigned long long* __restrict__ gkey) {
    __shared__ float tb[NOBJ_ * 4];
    __shared__ unsigned long long lkey[NOBJ_];
    const int b = blockIdx.y;
    const int tid = threadIdx.x;
    if (tid < NOBJ_) {
        const float* tr = tgt + ((size_t)b * NOBJ_ + tid) * 5;
        tb[tid * 4 + 0] = tr[0];
        tb[tid * 4 + 1] = tr[1];
        tb[tid * 4 + 2] = tr[2];
        tb[tid * 4 + 3] = tr[3];
        lkey[tid] = 0ull;
    }
    __syncthreads();
    const int p = blockIdx.x * blockDim.x + tid;
    const bool inb = p < P_;
    const int pc = inb ? p : 0;
    const int lane = tid & 31;
    const float ax = anc[4 * pc + 0], ay = anc[4 * pc + 1];
    const float ax2 = ax + anc[4 * pc + 2], ay2 = ay + anc[4 * pc + 3];
    const float areaA = (ax2 - ax) * (ay2 - ay);
    float bestO = -1.f;
    int bestJ = 0;
    for (int j = 0; j < NOBJ_; ++j) {
        float x1 = tb[4 * j], y1 = tb[4 * j + 1], x2 = tb[4 * j + 2], y2 = tb[4 * j + 3];
        float iw = fmaxf(fminf(x2, ax2) - fmaxf(x1, ax), 0.f);
        float ih = fmaxf(fminf(y2, ay2) - fmaxf(y1, ay), 0.f);
        float inter = iw * ih;
        float areaT = (x2 - x1) * (y2 - y1);
        float o = inter / (areaT + areaA - inter);
        if (inb && o > bestO) { bestO = o; bestJ = j; }   // strict > : first-occurrence tie (axis=0)
        unsigned long long key = inb ? packKey(o, p) : 0ull;  // ~p : smallest-p tie (axis=1)
        #pragma unroll
        for (int d = 16; d > 0; d >>= 1) {
            unsigned long long ok = __shfl_xor(key, d, 32);
            if (ok > key) key = ok;
        }
        if (lane == 0) atomicMax(&lkey[j], key);
    }
    if (inb) mt[(size_t)b * P_ + p] = (bestO > 0.5f) ? (signed char)(bestJ + 32)
                                                     : (signed char)(-1);
    __syncthreads();
    if (tid < NOBJ_) atomicMax(&gkey[(size_t)b * NOBJ_ + tid], lkey[tid]);
}

// ---------------- scatter the "first" (best-prior-per-truth) matches ----------------
__global__ void k_scatter(const unsigned long long* __restrict__ gkey,
                          signed char* __restrict__ mt) {
    const int b = blockIdx.x;
    if (threadIdx.x != 0) return;
    for (int j = 0; j < NOBJ_; ++j) {                // ascending: later j wins on duplicates
        unsigned long long k = gkey[(size_t)b * NOBJ_ + j];
        unsigned p = 0xFFFFFFFFu - (unsigned)(k & 0xFFFFFFFFu);
        signed char cur = mt[(size_t)b * P_ + p];
        if (cur < 32) mt[(size_t)b * P_ + p] = (signed char)j;  // over_t ("second", >=32) wins
    }
}

// ---------------- big streaming kernel: lse via WMMA, cls values, pos losses ----------------
// One wave handles 16 priors; lanes m and m+16 split a row: classes [0,40) / [40,80),
// class 80 handled once (half-1) via a branchless 0/1 mask. All loads are unconditional
// and held in registers, so the 21-WMMA chain runs with full EXEC and no divergence.
__global__ void __launch_bounds__(256) k_lse(const float* __restrict__ conf,
                                             const float* __restrict__ locp,
                                             const float* __restrict__ tgt,
                                             const float* __restrict__ anc,
                                             const signed char* __restrict__ mt,
                                             float* __restrict__ cls,
                                             float* __restrict__ acc,
                                             int* __restrict__ numPos) {
    const int wave = blockIdx.x * (blockDim.x >> 5) + (threadIdx.x >> 5);
    if (wave >= B_ * TILES_) return;                 // wave-uniform
    const int b = wave / TILES_;
    const int tile = wave - b * TILES_;
    const int lane = threadIdx.x & 31;
    const int m = lane & 15;
    const int hf = lane >> 4;                        // lanes m and m+16 share prior m
    const int p = tile * 16 + m;
    const bool valid = p < P_;
    const int pc = valid ? p : (P_ - 1);
    const float* row = conf + ((size_t)b * P_ + pc) * C_;
    __builtin_prefetch(row, 0, 3);
    const int c0 = hf ? 40 : 0;

    // unconditional loads: 40 consecutive classes per lane, kept in registers
    float vbuf[40];
    #pragma unroll
    for (int i = 0; i < 40; ++i) vbuf[i] = row[c0 + i];
    const float v80 = row[80];

    // row max: include v80 in both halves (union max unchanged) -> fully branchless
    float mx = v80;
    #pragma unroll
    for (int i = 0; i < 40; ++i) mx = fmaxf(mx, vbuf[i]);
    const float rowmax = fmaxf(mx, __shfl_xor(mx, 16, 32));

    // sum of exp on the matrix pipe: A = exp tile (16 priors x 4 classes), B = ones
    v8f accv = {0.f, 0.f, 0.f, 0.f, 0.f, 0.f, 0.f, 0.f};
    v2f ones;
    ones.x = 1.f;
    ones.y = 1.f;
    #pragma unroll
    for (int s = 0; s < 20; ++s) {
        v2f a;
        a.x = __expf(vbuf[2 * s] - rowmax);
        a.y = __expf(vbuf[2 * s + 1] - rowmax);
        accv = __builtin_amdgcn_wmma_f32_16x16x4_f32(false, a, false, ones,
                                                     (short)0, accv, false, false);
    }
    {   // class 80: counted once (half-1 lanes), mask by multiply (no branch)
        v2f a;
        a.x = (float)hf * __expf(v80 - rowmax);
        a.y = 0.f;
        accv = __builtin_amdgcn_wmma_f32_16x16x4_f32(false, a, false, ones,
                                                     (short)0, accv, false, false);
    }

    // extract row sum for prior m: element (M=m, N=0) -> acc[m%8] at lane 0 (m<8) or lane 16
    float rowsum = 0.f;
    #pragma unroll
    for (int j = 0; j < 8; ++j) {
        float a0 = __shfl(accv[j], 0, 32);
        float a16 = __shfl(accv[j], 16, 32);
        if ((m & 7) == j) rowsum = (m < 8) ? a0 : a16;
    }
    const float lse = rowmax + __logf(rowsum);
    const signed char mtv = mt[(size_t)b * P_ + pc];
    const bool pos = mtv >= 0;
    const float clsv = pos ? 0.f : (lse - vbuf[0]);  // vbuf[0] == row[0] on half-0 lanes
    if (valid && hf == 0) {
        cls[(size_t)b * P_ + p] = clsv;
        if (pos) {
            int t = (mtv >= 32) ? (mtv - 32) : (int)mtv;
            const float* tr = tgt + ((size_t)b * NOBJ_ + t) * 5;
            int lab = (int)tr[4];
            atomicAdd(&acc[1], lse - row[lab]);      // pos CE
            atomicAdd(&numPos[b], 1);
            // loc smooth-L1 vs encode(xyxy->xywh(truth), anchor)
            float ax = anc[4 * p], ay = anc[4 * p + 1], aw = anc[4 * p + 2], ah = anc[4 * p + 3];
            float tw = tr[2] - tr[0], th = tr[3] - tr[1];
            float g0 = (tr[0] - ax) / aw;
            float g1 = (tr[1] - ay) / ah;
            float g2 = __logf(tw) - __logf(aw);
            float g3 = __logf(th) - __logf(ah);
            const float* lp = locp + ((size_t)b * P_ + p) * 4;
            float s = 0.f, d;
            d = fabsf(lp[0] - g0); s += (d < 1.f) ? 0.5f * d * d : d - 0.5f;
            d = fabsf(lp[1] - g1); s += (d < 1.f) ? 0.5f * d * d : d - 0.5f;
            d = fabsf(lp[2] - g2); s += (d < 1.f) ? 0.5f * d * d : d - 0.5f;
            d = fabsf(lp[3] - g3); s += (d < 1.f) ? 0.5f * d * d : d - 0.5f;
            atomicAdd(&acc[0], s);
        }
    }
}

// ---------------- per-batch top-k radix select + ordered tie scan + CE sum ----------------
__global__ void __launch_bounds__(256) k_select(const float* __restrict__ clsAll,
                                                const signed char* __restrict__ mtAll,
                                                const int* __restrict__ numPos,
                                                float* __restrict__ acc) {
    const int b = blockIdx.x;
    const float* v = clsAll + (size_t)b * P_;
    const signed char* mt = mtAll + (size_t)b * P_;
    int k = numPos[b] * 3;
    if (k > P_ - 1) k = P_ - 1;
    const bool doSel = k > 0;
    __shared__ unsigned hist[256];
    __shared__ unsigned sPrefix;
    __shared__ int sK;
    __shared__ int sBase;
    __shared__ int wtot[8];
    __shared__ float wsum[8];
    const int tid = threadIdx.x;
    const int lane = tid & 31, wid = tid >> 5;
    if (tid == 0) { sPrefix = 0u; sK = k; sBase = 0; }
    __syncthreads();
    if (doSel) {
        // 4-pass MSB->LSB radix select of k-th largest (cls >= 0 so uint order == float order)
        for (int shift = 24; shift >= 0; shift -= 8) {
            hist[tid] = 0u;
            __syncthreads();
            unsigned pfx = sPrefix;
            unsigned hiMask = (shift == 24) ? 0u : (0xFFFFFFFFu << (shift + 8));
            for (int p = tid; p < P_; p += 256) {
                unsigned bits = __float_as_uint(v[p]);
                if ((bits & hiMask) == pfx) atomicAdd(&hist[(bits >> shift) & 255], 1u);
            }
            __syncthreads();
            if (tid == 0) {
                int kk = sK;
                unsigned run = 0;
                for (int d = 255; d >= 0; --d) {
                    unsigned h = hist[d];
                    if (run + h >= (unsigned)kk) {
                        sPrefix = pfx | ((unsigned)d << shift);
                        sK = kk - (int)run;
                        break;
                    }
                    run += h;
                }
            }
            __syncthreads();
        }
    }
    const unsigned Tb = sPrefix;
    const int rem = sK;   // of the ties at T, select first `rem` in index order (stable sort)
    float local = 0.f;
    for (int p0 = 0; p0 < P_; p0 += 256) {
        int p = p0 + tid;
        bool inb = p < P_;
        float val = inb ? v[p] : 0.f;
        unsigned bits = __float_as_uint(val);
        bool eq = doSel && inb && (bits == Tb);
        bool gt = doSel && inb && (bits > Tb);
        unsigned wm = (unsigned)__ballot(eq);
        int wpre = (lane == 0) ? 0 : __popc(wm & (0xFFFFFFFFu >> (32 - lane)));
        if (lane == 0) wtot[wid] = __popc(wm);
        __syncthreads();
        int wbase = 0;
        for (int i = 0; i < wid; ++i) wbase += wtot[i];
        int eqIdx = sBase + wbase + wpre;
        bool sel = gt || (eq && eqIdx < rem);
        bool pos = inb && (mt[p] >= 0);
        if (inb && !pos) local += sel ? val : LN81_;   // ce_neg == cls; unselected -> ln(81)
        __syncthreads();
        if (tid == 0) {
            int t = 0;
            for (int i = 0; i < 8; ++i) t += wtot[i];
            sBase += t;
        }
        __syncthreads();
    }
    #pragma unroll
    for (int d = 16; d > 0; d >>= 1) local += __shfl_down(local, d, 32);
    if (lane == 0) wsum[wid] = local;
    __syncthreads();
    if (tid == 0) {
        float t = 0.f;
        for (int i = 0; i < 8; ++i) t += wsum[i];
        atomicAdd(&acc[1], t);
    }
}

// ---------------- finalize ----------------
__global__ void k_finalize(const float* __restrict__ acc,
                           const int* __restrict__ numPos,
                           float* __restrict__ out) {
    if (threadIdx.x == 0 && blockIdx.x == 0) {
        int n = 0;
        for (int b = 0; b < B_; ++b) n += numPos[b];
        float N = (float)n;
        out[0] = acc[0] / N;
        out[1] = acc[1] / N;
    }
}

extern "C" void kernel_launch(void* const* d_in, const int* in_sizes, int n_in,
                              void* d_out, int out_size, void* d_ws, size_t ws_size,
                              hipStream_t stream) {
    const float* locp = (const float*)d_in[0];   // (B,P,4)
    const float* conf = (const float*)d_in[1];   // (B,P,81)
    const float* tgt  = (const float*)d_in[2];   // (B,32,5)
    const float* anc  = (const float*)d_in[3];   // (P,4)
    float* out = (float*)d_out;

    char* ws = (char*)d_ws;
    size_t off = 0;
    float* cls = (float*)(ws + off);              off += (size_t)B_ * P_ * sizeof(float);
    signed char* mt = (signed char*)(ws + off);   off += (size_t)B_ * P_;
    off = (off + 15) & ~(size_t)15;
    unsigned long long* gkey = (unsigned long long*)(ws + off); off += (size_t)B_ * NOBJ_ * 8;
    int* numPos = (int*)(ws + off);               off += (size_t)B_ * 4;
    float* acc = (float*)(ws + off);              off += 16;

    k_init<<<8, 256, 0, stream>>>(gkey, numPos, acc);
    dim3 g1((P_ + 255) / 256, B_);
    k_match<<<g1, 256, 0, stream>>>(tgt, anc, mt, gkey);
    k_scatter<<<B_, 32, 0, stream>>>(gkey, mt);
    k_lse<<<(B_ * TILES_) / 8, 256, 0, stream>>>(conf, locp, tgt, anc, mt, cls, acc, numPos);
    k_select<<<B_, 256, 0, stream>>>(cls, mt, numPos, acc);
    k_finalize<<<1, 32, 0, stream>>>(acc, numPos, out);
}